// DTODRLGATEncoder_64544768524831
// MI455X (gfx1250) — compile-verified
//
#include <hip/hip_runtime.h>
#include <hip/hip_bf16.h>

// ---------------------------------------------------------------------------
// GAT encoder (3 layers) for MI455X / gfx1250.
// Dense transforms: V_WMMA_F32_16X16X4_F32, one wave per 16x16 tile,
// fully unpredicated inner loop (padded operands, pre-transposed W).
// Edge-phase segment softmax + scatter: f32 global atomics.
// ---------------------------------------------------------------------------

typedef float v2f __attribute__((ext_vector_type(2)));
typedef float v8f __attribute__((ext_vector_type(8)));

#define NEG_SLOPE 0.2f
#define LDF 144   // padded leading dimension for node-feature matrices (mult of 4 & 16)

__device__ __forceinline__ float leaky(float x) { return x > 0.f ? x : NEG_SLOPE * x; }

// float atomic max via signed/unsigned integer ordering trick
__device__ __forceinline__ void atomicMaxF(float* addr, float val) {
    if (val >= 0.f) {
        atomicMax((int*)addr, __float_as_int(val));
    } else {
        atomicMin((unsigned int*)addr, __float_as_uint(val));
    }
}

// ---------------------------------------------------------------------------
// GEMM: Out[M, 16*tilesN] = X[M, KPAD] @ Wt^T   (Wt is [Npad][KPAD], padded)
// One wave per 16x16 tile. No bounds checks in the hot loop.
// M must be a multiple of 16; KPAD a multiple of 4; pads are zero-filled.
// ---------------------------------------------------------------------------
template <int KPAD>
__global__ void gemm_f32_wmma(const float* __restrict__ X, int ldX,
                              const float* __restrict__ Wt,
                              float* __restrict__ Out, int ldOut,
                              int M, int tilesN) {
    const int lane = threadIdx.x & 31;
    const int wave = threadIdx.x >> 5;
    const int tile = blockIdx.x * (blockDim.x >> 5) + wave;
    const int tilesM = M >> 4;
    if (tile >= tilesM * tilesN) return;            // wave-uniform exit

    const int mt = tile / tilesN;
    const int nt = tile % tilesN;
    const int half = lane >> 4;                     // 0: lanes 0-15, 1: 16-31
    const int l    = lane & 15;

    // A fragment: row = mt*16 + l, K pair starts at 2*half (+k0)
    const float* aptr = X  + (size_t)(mt * 16 + l) * ldX  + 2 * half;
    // B fragment (from transposed W): col = nt*16 + l, same K pair
    const float* bptr = Wt + (size_t)(nt * 16 + l) * KPAD + 2 * half;

    v8f acc = {};
    #pragma unroll 4
    for (int k0 = 0; k0 < KPAD; k0 += 4) {
        const v2f a = *(const v2f*)(aptr + k0);     // one b64 load
        const v2f b = *(const v2f*)(bptr + k0);     // one b64 load
        acc = __builtin_amdgcn_wmma_f32_16x16x4_f32(false, a, false, b,
                                                    (short)0, acc, false, false);
    }

    float* o = Out + (size_t)(mt * 16 + half * 8) * ldOut + nt * 16 + l;
    #pragma unroll
    for (int r = 0; r < 8; ++r) o[(size_t)r * ldOut] = acc[r];
}

// ---------------------------------------------------------------------------
// Padding helpers (one-time, tiny)
// ---------------------------------------------------------------------------
__global__ void pad_rows(const float* __restrict__ src, float* __restrict__ dst,
                         int Nrows, int K, int Kpad) {
    const long long total = (long long)Nrows * Kpad;
    for (long long i = blockIdx.x * (long long)blockDim.x + threadIdx.x; i < total;
         i += (long long)gridDim.x * blockDim.x) {
        const int c = (int)(i % Kpad);
        const long long r = i / Kpad;
        dst[i] = (c < K) ? src[r * K + c] : 0.f;
    }
}

// Wt[n*Kpad + k] = W[k*Nout + n] (zero padded)
__global__ void pad_w_transpose(const float* __restrict__ W, float* __restrict__ Wt,
                                int K, int Nout, int Kpad, int Npad) {
    const int i = blockIdx.x * blockDim.x + threadIdx.x;
    if (i >= Npad * Kpad) return;
    const int n = i / Kpad;
    const int k = i % Kpad;
    Wt[i] = (n < Nout && k < K) ? W[(size_t)k * Nout + n] : 0.f;
}

// ---------------------------------------------------------------------------
// Per-node attention logit components: al_s[n,h] = <h[n,h,:], a_src[h,:]>
// ---------------------------------------------------------------------------
__global__ void compute_al(const float* __restrict__ Hm, int ldT,
                           const float* __restrict__ a_s, const float* __restrict__ a_d,
                           float* __restrict__ alS, float* __restrict__ alD,
                           int N, int Hh, int F) {
    const int i = blockIdx.x * blockDim.x + threadIdx.x;
    if (i >= N * Hh) return;
    const int n = i / Hh, h = i % Hh;
    const float* row = Hm + (size_t)n * ldT + (size_t)h * F;
    float s = 0.f, d = 0.f;
    for (int f = 0; f < F; ++f) {
        const float v = row[f];
        s += v * a_s[h * F + f];
        d += v * a_d[h * F + f];
    }
    alS[i] = s;
    alD[i] = d;
}

__global__ void init_mx_den(float* __restrict__ mx, float* __restrict__ den, int count) {
    const int i = blockIdx.x * blockDim.x + threadIdx.x;
    if (i < count) { mx[i] = -__builtin_huge_valf(); den[i] = 0.f; }
}

__global__ void zero_f32(float* __restrict__ p, long long count) {
    for (long long i = blockIdx.x * (long long)blockDim.x + threadIdx.x; i < count;
         i += (long long)gridDim.x * blockDim.x)
        p[i] = 0.f;
}

// ---------------------------------------------------------------------------
// Edge pass 1: segment max of leaky(al_s[src] + al_d[dst]) over dst.
// One thread per edge, head loop unrolled. e >= E are self loops.
// ---------------------------------------------------------------------------
template <int HH>
__global__ void edge_max_k(const int* __restrict__ src, const int* __restrict__ dst,
                           long long E, long long Etot,
                           const float* __restrict__ alS, const float* __restrict__ alD,
                           float* __restrict__ mx) {
    for (long long e = blockIdx.x * (long long)blockDim.x + threadIdx.x; e < Etot;
         e += (long long)gridDim.x * blockDim.x) {
        const int s = (e < E) ? src[e] : (int)(e - E);
        const int d = (e < E) ? dst[e] : (int)(e - E);
        #pragma unroll
        for (int h = 0; h < HH; ++h) {
            const float lg = leaky(alS[(size_t)s * HH + h] + alD[(size_t)d * HH + h]);
            atomicMaxF(&mx[(size_t)d * HH + h], lg);
        }
    }
}

// ---------------------------------------------------------------------------
// Edge pass 2: ex = exp(logit - mx[dst]); den[dst] += ex; aEx[e,h] = ex.
// ---------------------------------------------------------------------------
template <int HH>
__global__ void edge_expsum_k(const int* __restrict__ src, const int* __restrict__ dst,
                              long long E, long long Etot,
                              const float* __restrict__ alS, const float* __restrict__ alD,
                              const float* __restrict__ mx, float* __restrict__ den,
                              float* __restrict__ aEx) {
    for (long long e = blockIdx.x * (long long)blockDim.x + threadIdx.x; e < Etot;
         e += (long long)gridDim.x * blockDim.x) {
        const int s = (e < E) ? src[e] : (int)(e - E);
        const int d = (e < E) ? dst[e] : (int)(e - E);
        #pragma unroll
        for (int h = 0; h < HH; ++h) {
            const float lg = leaky(alS[(size_t)s * HH + h] + alD[(size_t)d * HH + h]);
            const float ex = __expf(lg - mx[(size_t)d * HH + h]);
            aEx[e * HH + h] = ex;
            atomicAdd(&den[(size_t)d * HH + h], ex);
        }
    }
}

// ---------------------------------------------------------------------------
// Edge pass 3: alpha = ex / (den[dst] + 1e-16)
// ---------------------------------------------------------------------------
template <int HH>
__global__ void finalize_alpha_k(const int* __restrict__ dst, long long E, long long Etot,
                                 const float* __restrict__ den, float* __restrict__ aEx) {
    for (long long e = blockIdx.x * (long long)blockDim.x + threadIdx.x; e < Etot;
         e += (long long)gridDim.x * blockDim.x) {
        const int d = (e < E) ? dst[e] : (int)(e - E);
        #pragma unroll
        for (int h = 0; h < HH; ++h)
            aEx[e * HH + h] = aEx[e * HH + h] / (den[(size_t)d * HH + h] + 1e-16f);
    }
}

// ---------------------------------------------------------------------------
// Edge pass 4: out[dst, c] += T[src, c] * alpha[e, c/F]
// ---------------------------------------------------------------------------
__global__ void edge_scatter(const int* __restrict__ src, const int* __restrict__ dst,
                             long long E, long long Etot,
                             const float* __restrict__ alpha,
                             const float* __restrict__ Hm, int ldT,
                             float* __restrict__ out, int ldU,
                             int Hh, int F) {
    const int HF = Hh * F;
    const long long total = Etot * HF;
    for (long long i = blockIdx.x * (long long)blockDim.x + threadIdx.x; i < total;
         i += (long long)gridDim.x * blockDim.x) {
        const long long e = i / HF;
        const int c = (int)(i % HF);
        const int h = c / F;
        const int s = (e < E) ? src[e] : (int)(e - E);
        const int d = (e < E) ? dst[e] : (int)(e - E);
        const float a = alpha[e * Hh + h];
        atomicAdd(&out[(size_t)d * ldU + c], Hm[(size_t)s * ldT + c] * a);
    }
}

// ---------------------------------------------------------------------------
// out[n, c] = (leaky?) leaky_relu(out[n, c] + bias[c]), c < HF (pads untouched)
// ---------------------------------------------------------------------------
__global__ void bias_act(float* __restrict__ out, int ldU, const float* __restrict__ b,
                         int N, int HF, int doLeaky) {
    const long long total = (long long)N * HF;
    for (long long i = blockIdx.x * (long long)blockDim.x + threadIdx.x; i < total;
         i += (long long)gridDim.x * blockDim.x) {
        const long long n = i / HF;
        const int c = (int)(i % HF);
        float v = out[n * ldU + c] + b[c];
        if (doLeaky) v = leaky(v);
        out[n * ldU + c] = v;
    }
}

// ---------------------------------------------------------------------------
// Host orchestration
// ---------------------------------------------------------------------------
struct LayerBufs {
    const int* src; const int* dst; long long E; int N;
    float *alS, *alD, *mx, *den, *aEx;
};

template <int KPAD, int HH>
static void run_gat_layer(const float* X, int ldX, const float* Wt, int tilesN,
                          const float* aS, const float* aD, const float* bias,
                          int F, int doLeaky,
                          float* T, int ldT, float* U, int ldU,
                          const LayerBufs& lb, hipStream_t stream) {
    const int HF = HH * F;
    const int N = lb.N;
    const long long E = lb.E;
    const long long Etot = E + (long long)N;

    // 1) dense transform via WMMA
    {
        const int tiles = (N >> 4) * tilesN;
        const int blocks = (tiles + 7) / 8;          // 8 waves per 256-thread block
        gemm_f32_wmma<KPAD><<<blocks, 256, 0, stream>>>(X, ldX, Wt, T, ldT, N, tilesN);
    }
    // 2) attention logit components + stat init
    {
        const int cnt = N * HH;
        compute_al<<<(cnt + 255) / 256, 256, 0, stream>>>(T, ldT, aS, aD,
                                                          lb.alS, lb.alD, N, HH, F);
        init_mx_den<<<(cnt + 255) / 256, 256, 0, stream>>>(lb.mx, lb.den, cnt);
    }
    // 3) segment softmax over edges
    edge_max_k<HH><<<4096, 256, 0, stream>>>(lb.src, lb.dst, E, Etot,
                                             lb.alS, lb.alD, lb.mx);
    edge_expsum_k<HH><<<4096, 256, 0, stream>>>(lb.src, lb.dst, E, Etot,
                                                lb.alS, lb.alD, lb.mx, lb.den, lb.aEx);
    finalize_alpha_k<HH><<<4096, 256, 0, stream>>>(lb.dst, E, Etot, lb.den, lb.aEx);
    // 4) aggregate (zero whole padded buffer so pad cols stay 0 for next layer)
    zero_f32<<<2048, 256, 0, stream>>>(U, (long long)N * ldU);
    edge_scatter<<<8192, 256, 0, stream>>>(lb.src, lb.dst, E, Etot, lb.aEx,
                                           T, ldT, U, ldU, HH, F);
    // 5) bias + activation (only real columns)
    bias_act<<<2048, 256, 0, stream>>>(U, ldU, bias, N, HF, doLeaky);
}

extern "C" void kernel_launch(void* const* d_in, const int* in_sizes, int n_in,
                              void* d_out, int out_size, void* d_ws, size_t ws_size,
                              hipStream_t stream) {
    const float* x      = (const float*)d_in[0];
    const int*   eidx   = (const int*)d_in[1];
    const float* W0     = (const float*)d_in[2];
    const float* a_src0 = (const float*)d_in[3];
    const float* a_dst0 = (const float*)d_in[4];
    const float* b0     = (const float*)d_in[5];
    const float* W1     = (const float*)d_in[6];
    const float* a_src1 = (const float*)d_in[7];
    const float* a_dst1 = (const float*)d_in[8];
    const float* b1     = (const float*)d_in[9];
    const float* W2     = (const float*)d_in[10];
    const float* a_src2 = (const float*)d_in[11];
    const float* a_dst2 = (const float*)d_in[12];
    const float* b2     = (const float*)d_in[13];

    const int N       = in_sizes[0] / 6;
    const long long E = in_sizes[1] / 2;
    const int* src = eidx;
    const int* dst = eidx + E;

    // workspace layout (floats)
    float* p   = (float*)d_ws;
    float* xp  = p; p += (size_t)N * 8;         // layer-1 input padded N x 8
    float* ws0 = p; p += (size_t)N * LDF;       // transform buffer T, ld = 144
    float* ws1 = p; p += (size_t)N * LDF;       // feature ping-pong,  ld = 144
    float* alS = p; p += (size_t)N * 3;
    float* alD = p; p += (size_t)N * 3;
    float* mx  = p; p += (size_t)N * 3;
    float* den = p; p += (size_t)N * 3;
    float* aEx = p; p += (E + (size_t)N) * 3;
    float* W0t = p; p += (size_t)LDF * 8;       // [144][8]
    float* W1t = p; p += (size_t)LDF * LDF;     // [144][144]
    float* W2t = p; p += (size_t)LDF * LDF;     // [144][144]

    // one-time padding / transposition
    pad_rows<<<512, 256, 0, stream>>>(x, xp, N, 6, 8);
    pad_w_transpose<<<(LDF * 8   + 255) / 256, 256, 0, stream>>>(W0, W0t, 6,   129, 8,   LDF);
    pad_w_transpose<<<(LDF * LDF + 255) / 256, 256, 0, stream>>>(W1, W1t, 129, 129, LDF, LDF);
    pad_w_transpose<<<(LDF * LDF + 255) / 256, 256, 0, stream>>>(W2, W2t, 129, 128, LDF, LDF);

    LayerBufs lb{src, dst, E, N, alS, alD, mx, den, aEx};

    // Layer 1: xp[N,8] -> ws1[N,129 (ld144)], leaky     (tilesN = 144/16 = 9)
    run_gat_layer<8, 3>(xp, 8, W0t, 9, a_src0, a_dst0, b0, 43, 1,
                        ws0, LDF, ws1, LDF, lb, stream);
    // Layer 2: ws1 -> ws1 (U aliases X; X only read by GEMM), leaky
    run_gat_layer<LDF, 3>(ws1, LDF, W1t, 9, a_src1, a_dst1, b1, 43, 1,
                          ws0, LDF, ws1, LDF, lb, stream);
    // Layer 3: ws1 -> d_out[N,128], single head, no activation (tilesN = 128/16 = 8)
    run_gat_layer<LDF, 1>(ws1, LDF, W2t, 8, a_src2, a_dst2, b2, 128, 0,
                          ws0, LDF, (float*)d_out, 128, lb, stream);
}